// ARKIModule_62019327754835
// MI455X (gfx1250) — compile-verified
//
#include <hip/hip_runtime.h>
#include <math.h>

#define H_IMG 1440
#define W_IMG 1920
#define NPIX (H_IMG * W_IMG)
static __device__ __constant__ float kEPS = 1e-8f;

typedef float v2f __attribute__((ext_vector_type(2)));
typedef float v8f __attribute__((ext_vector_type(8)));

#define TIN_ROWS 32
#define TIN_COLS 24          // 22 valid extended cols, padded to 6 K-chunks of 4
#define TIN_ELEMS (TIN_ROWS * TIN_COLS)
#define HS_STRIDE 17         // 16 cols + pad (LDS bank spread)

// -------------------------------------------------------------------------
// K1: tiny per-launch precompute (rotations, intrinsics products, accum zero)
// params layout (floats): [0]=f [1]=cx [2]=cy [3..11]=R1 row-major
//   view t in {0,1}: base=12+t*12 : A(9 row-major) Cd(3)
//   [40]=score accum  [41]=pixel accum
// -------------------------------------------------------------------------
__global__ void k1_precompute(const float* focal, const float* aa,
                              const float* centers, float* params) {
    float f  = focal[0];
    float cx = W_IMG * 0.5f, cy = H_IMG * 0.5f;
    float R[3][3][3];
    for (int c = 0; c < 3; ++c) {
        float vx = aa[c*3+0], vy = aa[c*3+1], vz = aa[c*3+2];
        float th = sqrtf(vx*vx + vy*vy + vz*vz) + 1e-12f;
        float kx = vx/th, ky = vy/th, kz = vz/th;
        float S[3][3] = {{0.f,-kz, ky},{ kz,0.f,-kx},{-ky, kx,0.f}};
        float S2[3][3];
        for (int i = 0; i < 3; ++i)
            for (int j = 0; j < 3; ++j) {
                float s = 0.f;
                for (int k = 0; k < 3; ++k) s += S[i][k]*S[k][j];
                S2[i][j] = s;
            }
        float sn = sinf(th), cc = 1.f - cosf(th);
        for (int i = 0; i < 3; ++i)
            for (int j = 0; j < 3; ++j)
                R[c][i][j] = (i==j ? 1.f : 0.f) + sn*S[i][j] + cc*S2[i][j];
    }
    float K[3][3] = {{f,0.f,cx-0.5f},{0.f,f,cy-0.5f},{0.f,0.f,1.f}};
    params[0] = f; params[1] = cx; params[2] = cy;
    for (int i = 0; i < 3; ++i)
        for (int j = 0; j < 3; ++j) params[3 + i*3 + j] = R[0][i][j];
    for (int t = 0; t < 2; ++t) {
        float* base = params + 12 + t*12;
        for (int i = 0; i < 3; ++i)
            for (int j = 0; j < 3; ++j) {
                float s = 0.f;
                for (int k = 0; k < 3; ++k) s += K[i][k]*R[t+1][k][j];
                base[i*3 + j] = s;
            }
        for (int j = 0; j < 3; ++j) base[9 + j] = centers[j] - centers[(t+1)*3 + j];
    }
    params[40] = 0.f;
    params[41] = 0.f;
}

// -------------------------------------------------------------------------
// K2: per-pixel homography warp + bilinear sample of target image
// -------------------------------------------------------------------------
__global__ void k2_warp(const float* __restrict__ depth,
                        const float* __restrict__ tgts,
                        const float* __restrict__ params,
                        float* __restrict__ warped) {
    int idx = blockIdx.x * blockDim.x + threadIdx.x;
    int t   = blockIdx.y;
    if (idx >= NPIX) return;
    int y = idx / W_IMG, x = idx - y * W_IMG;

    float f = params[0], cx = params[1], cy = params[2];
    float d  = depth[idx];
    float dl = (x > 0)        ? depth[idx - 1]     : 0.f;
    float dr = (x < W_IMG-1)  ? depth[idx + 1]     : 0.f;
    float dtp= (y > 0)        ? depth[idx - W_IMG] : 0.f;
    float dbt= (y < H_IMG-1)  ? depth[idx + W_IMG] : 0.f;
    float du = 0.5f * (dr - dl);
    float dv = 0.5f * (dbt - dtp);

    float nx = f * du, ny = f * dv;
    float nz = (cx - (float)x)*du + (cy - (float)y)*dv - d;
    float nn = sqrtf(nx*nx + ny*ny + nz*nz) + kEPS;
    float inv = 1.f / (nn * (d + kEPS));
    float ndx = nx*inv, ndy = ny*inv, ndz = nz*inv;

    float px = ((float)x - (cx - 0.5f)) / f;
    float py = ((float)y - (cy - 0.5f)) / f;
    float s  = ndx*px + ndy*py + ndz;

    const float* R1 = params + 3;
    const float* A  = params + 12 + t*12;
    const float* Cd = A + 9;
    float vx = R1[0]*px + R1[3]*py + R1[6] + s*Cd[0];
    float vy = R1[1]*px + R1[4]*py + R1[7] + s*Cd[1];
    float vz = R1[2]*px + R1[5]*py + R1[8] + s*Cd[2];
    float wx = A[0]*vx + A[1]*vy + A[2]*vz;
    float wy = A[3]*vx + A[4]*vy + A[5]*vz;
    float wz = A[6]*vx + A[7]*vy + A[8]*vz + kEPS;
    float sx = wx / wz, sy = wy / wz;

    const float* img = tgts + (size_t)t * NPIX;
    float x0 = floorf(sx), y0 = floorf(sy);
    float wx1 = sx - x0, wy1 = sy - y0;
    auto tap = [&](float xi, float yi) -> float {
        bool valid = (xi >= 0.f) && (xi <= (float)(W_IMG-1)) &&
                     (yi >= 0.f) && (yi <= (float)(H_IMG-1));
        int xc = (int)fminf(fmaxf(xi, 0.f), (float)(W_IMG-1));
        int yc = (int)fminf(fmaxf(yi, 0.f), (float)(H_IMG-1));
        return valid ? img[yc * W_IMG + xc] : 0.f;
    };
    float val = (1.f-wy1)*(1.f-wx1)*tap(x0,     y0    )
              + (1.f-wy1)*     wx1 *tap(x0+1.f, y0    )
              +      wy1 *(1.f-wx1)*tap(x0,     y0+1.f)
              +      wy1 *     wx1 *tap(x0+1.f, y0+1.f);
    warped[(size_t)t * NPIX + idx] = val;
}

// -------------------------------------------------------------------------
// 7x7 zero-padded box MEAN of a 22x22 extended tile (in LDS, zero-padded to
// 32x24) -> 16x16 outputs in D-fragment layout. Separable, via
// V_WMMA_F32_16X16X4_F32 with in-register banded 0/1 weight matrices.
// One wave32 per tile; EXEC all ones at every WMMA (uniform control flow).
// -------------------------------------------------------------------------
__device__ __forceinline__ v8f box_filter_tile(const float* tin, float* hs) {
    int lane = threadIdx.x;        // 0..31
    int m    = lane & 15;          // A-row / B,C,D column
    int ksub = (lane >> 4) << 1;   // K sub-offset per lane half
    __syncthreads();               // hs reuse guard
    // ---- horizontal: H(16x16/grp) = X(16x24) * W(24x16), W[k][c]=1 iff c<=k<=c+6
    for (int grp = 0; grp < 2; ++grp) {
        v8f acc = {};
        int row = grp * 16 + m;
#pragma unroll
        for (int kc = 0; kc < 6; ++kc) {
            int k0 = kc * 4 + ksub;
            v2f a, b;
            a.x = tin[row * TIN_COLS + k0];
            a.y = tin[row * TIN_COLS + k0 + 1];
            b.x = (k0     >= m && k0     <= m + 6) ? 1.f : 0.f;
            b.y = (k0 + 1 >= m && k0 + 1 <= m + 6) ? 1.f : 0.f;
            acc = __builtin_amdgcn_wmma_f32_16x16x4_f32(
                false, a, false, b, (short)0, acc, false, false);
        }
        int rbase = grp * 16 + ((lane >> 4) << 3);
#pragma unroll
        for (int v = 0; v < 8; ++v)
            hs[(rbase + v) * HS_STRIDE + m] = acc[v];
    }
    __syncthreads();
    // ---- vertical: D(16x16) = V(16x24) * H(24x16), V[r][k]=1/49 iff r<=k<=r+6
    v8f acc = {};
#pragma unroll
    for (int kc = 0; kc < 6; ++kc) {
        int k0 = kc * 4 + ksub;
        v2f a, b;
        a.x = (k0     >= m && k0     <= m + 6) ? (1.f / 49.f) : 0.f;
        a.y = (k0 + 1 >= m && k0 + 1 <= m + 6) ? (1.f / 49.f) : 0.f;
        b.x = hs[(k0    ) * HS_STRIDE + m];
        b.y = hs[(k0 + 1) * HS_STRIDE + m];
        acc = __builtin_amdgcn_wmma_f32_16x16x4_f32(
            false, a, false, b, (short)0, acc, false, false);
    }
    return acc;
}

// -------------------------------------------------------------------------
// K3: box means: z=0 -> mr from ref, z=1,2 -> mt[z-1] from warped[z-1]
// -------------------------------------------------------------------------
__global__ void k3_boxmean(const float* __restrict__ ref,
                           const float* __restrict__ warped,
                           float* __restrict__ mr, float* __restrict__ mt) {
    __shared__ float tin[TIN_ELEMS];
    __shared__ float hs[32 * HS_STRIDE];
    int tz = blockIdx.z;
    const float* src = (tz == 0) ? ref : warped + (size_t)(tz-1) * NPIX;
    float*       dst = (tz == 0) ? mr  : mt     + (size_t)(tz-1) * NPIX;
    int gx0 = blockIdx.x * 16, gy0 = blockIdx.y * 16, lane = threadIdx.x;

    for (int i = lane; i < TIN_ELEMS; i += 32) tin[i] = 0.f;
    __syncthreads();
    for (int i = lane; i < 22 * 22; i += 32) {
        int r = i / 22, c = i - r * 22;
        int gy = gy0 - 3 + r, gx = gx0 - 3 + c;
        float v = 0.f;
        if (gy >= 0 && gy < H_IMG && gx >= 0 && gx < W_IMG)
            v = src[gy * W_IMG + gx];
        tin[r * TIN_COLS + c] = v;
    }
    __syncthreads();
    v8f mean = box_filter_tile(tin, hs);
    int col = lane & 15, rb = (lane >> 4) << 3;
#pragma unroll
    for (int v = 0; v < 8; ++v)
        dst[(gy0 + rb + v) * W_IMG + gx0 + col] = mean[v];
}

// -------------------------------------------------------------------------
// K4: ZNCC tiles + global reduction. One wave32 per 16x16 tile per view.
// -------------------------------------------------------------------------
__global__ void k4_zncc(const float* __restrict__ ref,
                        const float* __restrict__ warped,
                        const float* __restrict__ mr,
                        const float* __restrict__ mt,
                        float* __restrict__ params) {
    __shared__ float tinR[TIN_ELEMS];
    __shared__ float tinT[TIN_ELEMS];
    __shared__ float tinP[TIN_ELEMS];
    __shared__ float hs[32 * HS_STRIDE];
    int t = blockIdx.z;
    const float* wmap = warped + (size_t)t * NPIX;
    const float* mtm  = mt     + (size_t)t * NPIX;
    int gx0 = blockIdx.x * 16, gy0 = blockIdx.y * 16, lane = threadIdx.x;

    for (int i = lane; i < TIN_ELEMS; i += 32) { tinR[i] = 0.f; tinT[i] = 0.f; }
    __syncthreads();
    for (int i = lane; i < 22 * 22; i += 32) {
        int r = i / 22, c = i - r * 22;
        int gy = gy0 - 3 + r, gx = gx0 - 3 + c;
        float rc = 0.f, tc = 0.f;
        if (gy >= 0 && gy < H_IMG && gx >= 0 && gx < W_IMG) {
            int p = gy * W_IMG + gx;
            rc = ref[p]  - mr[p];
            tc = wmap[p] - mtm[p];
        }
        tinR[r * TIN_COLS + c] = rc;
        tinT[r * TIN_COLS + c] = tc;
    }
    __syncthreads();

    for (int i = lane; i < TIN_ELEMS; i += 32) tinP[i] = tinR[i] * tinT[i];
    v8f cov  = box_filter_tile(tinP, hs);
    __syncthreads();
    for (int i = lane; i < TIN_ELEMS; i += 32) tinP[i] = tinR[i] * tinR[i];
    v8f varR = box_filter_tile(tinP, hs);
    __syncthreads();
    for (int i = lane; i < TIN_ELEMS; i += 32) tinP[i] = tinT[i] * tinT[i];
    v8f varT = box_filter_tile(tinP, hs);

    float s = 0.f, cnt = 0.f;
    int col = lane & 15, rb = (lane >> 4) << 3;
#pragma unroll
    for (int v = 0; v < 8; ++v) {
        int gy = gy0 + rb + v, gx = gx0 + col;
        float wv = wmap[gy * W_IMG + gx];
        float valid = (wv != 0.f) ? 1.f : 0.f;
        float zm = cov[v] / sqrtf(varR[v] * varT[v] + kEPS);
        s   += zm * valid;
        cnt += valid;
    }
    for (int off = 16; off > 0; off >>= 1) {
        s   += __shfl_xor(s,   off, 32);
        cnt += __shfl_xor(cnt, off, 32);
    }
    if (lane == 0) {
        atomicAdd(&params[40], s);
        atomicAdd(&params[41], cnt);
    }
}

// -------------------------------------------------------------------------
// K5: finalize scalar loss
// -------------------------------------------------------------------------
__global__ void k5_finalize(const float* __restrict__ params,
                            float* __restrict__ out) {
    float score = params[40], pix = params[41];
    float mean  = score / fmaxf(pix, 1.f);
    out[0] = (pix > 0.f) ? 0.5f * (1.f - mean) : 0.f;
}

// -------------------------------------------------------------------------
extern "C" void kernel_launch(void* const* d_in, const int* in_sizes, int n_in,
                              void* d_out, int out_size, void* d_ws, size_t ws_size,
                              hipStream_t stream) {
    const float* focal   = (const float*)d_in[0];
    const float* aa      = (const float*)d_in[1];
    const float* centers = (const float*)d_in[2];
    const float* ref     = (const float*)d_in[3];
    const float* depth   = (const float*)d_in[4];
    const float* tgts    = (const float*)d_in[5];
    float* out = (float*)d_out;

    size_t need = (size_t)(64 + 5 * NPIX) * sizeof(float);
    if (ws_size < need) return;   // scratch too small; cannot run

    float* ws     = (float*)d_ws;
    float* params = ws;
    float* warped = ws + 64;
    float* mr     = warped + 2 * (size_t)NPIX;
    float* mt     = mr + (size_t)NPIX;

    k1_precompute<<<1, 1, 0, stream>>>(focal, aa, centers, params);

    dim3 g2((NPIX + 255) / 256, 2);
    k2_warp<<<g2, 256, 0, stream>>>(depth, tgts, params, warped);

    dim3 g3(W_IMG / 16, H_IMG / 16, 3);
    k3_boxmean<<<g3, 32, 0, stream>>>(ref, warped, mr, mt);

    dim3 g4(W_IMG / 16, H_IMG / 16, 2);
    k4_zncc<<<g4, 32, 0, stream>>>(ref, warped, mr, mt, params);

    k5_finalize<<<1, 1, 0, stream>>>(params, out);
}